// LoRAConnector_12790412607982
// MI455X (gfx1250) — compile-verified
//
#include <hip/hip_runtime.h>
#include <cstdint>
#include <cstddef>

// Problem constants (match reference)
#define B_  4
#define T_  2048
#define N_  4
#define C_  1024
#define NC_ 4096          // N_*C_
#define A_  8
#define SINK_ITERS 20
#define EPS_ 1e-5f
#define JPACK 32          // packed projection columns: 4 pre + 4 post + 16 res + 8 pad

// LDS staging for the WMMA tile
#define CHUNK   128               // K elements per double-buffered stage
#define NCHUNK  (NC_ / CHUNK)     // 32
#define XPITCH  132               // padded x row pitch  -> conflict-free A reads
#define PPITCH  36                // padded phi row pitch (144B, 16B-aligned rows)

typedef float v2f __attribute__((ext_vector_type(2)));
typedef float v8f __attribute__((ext_vector_type(8)));
typedef int   v4i_t __attribute__((vector_size(16)));   // b128 payload type

// ---- gfx1250 async copy to LDS (ASYNCcnt path), with safe fallback --------
#if defined(__has_builtin)
# if __has_builtin(__builtin_amdgcn_global_load_async_to_lds_b128)
#  define USE_ASYNC 1
# endif
#endif
#ifndef USE_ASYNC
# define USE_ASYNC 0
#endif

#if USE_ASYNC
# define GPTR16(p) ((__attribute__((address_space(1))) v4i_t*)(p))
# define LPTR16(p) ((__attribute__((address_space(3))) v4i_t*)(p))
# define ASYNC_CP16(g, l) \
    __builtin_amdgcn_global_load_async_to_lds_b128(GPTR16(g), LPTR16(l), 0, 0)
# if __has_builtin(__builtin_amdgcn_s_wait_asynccnt)
#  define WAIT_ASYNC(n) __builtin_amdgcn_s_wait_asynccnt(n)
# else
#  define WAIT_ASYNC(n) asm volatile("s_wait_asynccnt %0" :: "i"(n) : "memory")
# endif
#else
# define ASYNC_CP16(g, l) (*(float4*)(l) = *(const float4*)(g))
# define WAIT_ASYNC(n) ((void)0)
#endif

// ---------------------------------------------------------------------------
// Kernel 1: pack phi_pre/phi_post/phi_res into phiP[a][d][32] (row-major),
// so the WMMA B-fragment feed in kernel 2 is one contiguous streamable array.
// ---------------------------------------------------------------------------
__global__ void __launch_bounds__(256)
pack_phi_kernel(const float* __restrict__ phi_pre,
                const float* __restrict__ phi_post,
                const float* __restrict__ phi_res,
                float* __restrict__ phiP) {
    int gid = blockIdx.x * blockDim.x + threadIdx.x;      // over A_*NC_*JPACK
    if (gid >= A_ * NC_ * JPACK) return;
    int j  = gid & (JPACK - 1);
    int ad = gid >> 5;                                    // a*NC_ + d
    int d  = ad & (NC_ - 1);
    int a  = ad >> 12;                                    // NC_ = 2^12
    float v = 0.0f;
    if (j < 4)        v = phi_pre [((size_t)a * NC_ + d) * 4  + j];
    else if (j < 8)   v = phi_post[((size_t)a * NC_ + d) * 4  + (j - 4)];
    else if (j < 24)  v = phi_res [((size_t)a * NC_ + d) * 16 + (j - 8)];
    phiP[gid] = v;
}

// ---------------------------------------------------------------------------
// Kernel 2: one wave32 per 16-token tile.
//   - x tile, phi chunk and input_norm_weight row all staged to LDS via
//     coalesced GLOBAL_LOAD_ASYNC_TO_LDS_B128, double-buffered (ASYNCcnt)
//   - (w .* x) @ phiP accumulated with V_WMMA_F32_16X16X4_F32 (2 col-tiles)
//   - sum(x^2) accumulated alongside; logits scaled by 1/rms at the end
//   - per-token tail: sigmoids + 20-iter Sinkhorn on exp(4x4) -> 24 coefs
// ---------------------------------------------------------------------------
__global__ void __launch_bounds__(32)
coef_kernel(const float* __restrict__ x,
            const float* __restrict__ w_all,      // input_norm_weight [A][NC]
            const float* __restrict__ phiP,       // packed [A][NC][32]
            const float* __restrict__ b_pre,      // [A][4]
            const float* __restrict__ b_post,     // [A][4]
            const float* __restrict__ b_res,      // [A][4][4]
            const float* __restrict__ alpha_pre,  // [A]
            const float* __restrict__ alpha_post, // [A]
            const float* __restrict__ alpha_res,  // [A]
            const int*   __restrict__ adapter_indices,  // [B]
            float* __restrict__ coef)             // [B*T][24]
{
    __shared__ __align__(16) float xs[2][16 * XPITCH];     // x tile (dbl buf)
    __shared__ __align__(16) float ps[2][CHUNK * PPITCH];  // phi chunk (dbl buf)
    __shared__ __align__(16) float wsh[NC_];               // norm-weight row
    __shared__ float Lg[16][33];
    __shared__ float rmsS[16];

    const int tile = blockIdx.x;            // 0 .. B_*T_/16 - 1
    const int b    = tile >> 7;             // T_/16 = 128 tiles per batch
    const int t0   = (tile & 127) << 4;
    const int lane = threadIdx.x;           // wave32
    const int half = lane >> 4;             // 0: K={k,k+1}, 1: K={k+2,k+3}
    const int n    = lane & 15;             // token row (A) / output col (B)
    const int idx  = adapter_indices[b];

    const float* xtile = x     + (size_t)(b * T_ + t0) * NC_;
    const float* wrow  = w_all + (size_t)idx * NC_;
    const float* prow  = phiP  + (size_t)idx * NC_ * JPACK;

    // Stage one x chunk: 16 rows x 128 floats; each op moves a fully
    // coalesced 512B row-segment (32 lanes x 16B). 16 async ops per chunk.
#define STAGE_X(cc) do {                                                   \
        const int   kb_  = (cc) * CHUNK;                                   \
        float*      dst_ = xs[(cc) & 1];                                   \
        _Pragma("unroll")                                                  \
        for (int j_ = 0; j_ < 16; ++j_) {                                  \
            const float* g_ = xtile + (size_t)j_ * NC_ + kb_ + lane * 4;   \
            float*       l_ = dst_ + j_ * XPITCH + lane * 4;               \
            ASYNC_CP16(g_, l_);                                            \
        } } while (0)

    // Stage one phi chunk: 128 rows x 32 floats (16KB); each op moves 512B
    // (4 consecutive 128B rows) into a pitch-36 LDS layout. 32 ops per chunk.
#define STAGE_P(cc) do {                                                   \
        const int   kb_  = (cc) * CHUNK;                                   \
        float*      dst_ = ps[(cc) & 1];                                   \
        const int   r_   = lane >> 3;          /* 0..3 row within group  */ \
        const int   col_ = (lane & 7) * 4;     /* 0..28 float column     */ \
        _Pragma("unroll")                                                  \
        for (int it_ = 0; it_ < 32; ++it_) {                               \
            const int row_ = it_ * 4 + r_;                                 \
            const float* g_ = prow + (size_t)(kb_ + row_) * JPACK + col_;  \
            float*       l_ = dst_ + row_ * PPITCH + col_;                 \
            ASYNC_CP16(g_, l_);                                            \
        } } while (0)

    // Stage the 16KB weight row: 32 async ops.
#pragma unroll
    for (int it = 0; it < NC_ / 128; ++it) {
        const float* g = wrow + it * 128 + lane * 4;
        float*       l = &wsh[it * 128 + lane * 4];
        ASYNC_CP16(g, l);
    }
    STAGE_X(0);  STAGE_P(0);
    STAGE_X(1);  STAGE_P(1);

    v8f acc0 = {};      // output cols 0..15  (pre 0-3, post 4-7, res 8-15)
    v8f acc1 = {};      // output cols 16..31 (res 16-23, zero pad 24-31)
    float ss = 0.0f;
    const int koff = half * 2;

    for (int c = 0; c < NCHUNK; ++c) {
        // 48 async ops per chunk; completions are in-order, so asynccnt<=48
        // with chunks c+1 (and c+2) in flight guarantees chunk c has landed.
        if (c + 2 < NCHUNK)      { STAGE_X(c + 2); STAGE_P(c + 2); WAIT_ASYNC(48); }
        else if (c + 1 < NCHUNK) { WAIT_ASYNC(48); }
        else                     { WAIT_ASYNC(0);  }

        const float* xbuf = xs[c & 1];
        const float* pbuf = ps[c & 1];
        const int    kb   = c * CHUNK;
#pragma unroll 4
        for (int k = 0; k < CHUNK; k += 4) {
            const int kl = k + koff;               // K within chunk (this half)
            const int kk = kb + kl;                // global K
            // A fragment from LDS: pitch 132 => banks 4n..4n+3, conflict-free
            float2 xv = *(const float2*)(xbuf + n * XPITCH + kl);
            float2 wv = *(const float2*)(&wsh[kk]);   // broadcast read
            ss += xv.x * xv.x + xv.y * xv.y;

            v2f a;  a[0] = xv.x * wv.x;  a[1] = xv.y * wv.y;
            // B fragments from LDS (pitch 36)
            v2f b0, b1;
            b0[0] = pbuf[kl       * PPITCH + n];
            b0[1] = pbuf[(kl + 1) * PPITCH + n];
            b1[0] = pbuf[kl       * PPITCH + n + 16];
            b1[1] = pbuf[(kl + 1) * PPITCH + n + 16];

            acc0 = __builtin_amdgcn_wmma_f32_16x16x4_f32(
                       false, a, false, b0, (short)0, acc0, false, false);
            acc1 = __builtin_amdgcn_wmma_f32_16x16x4_f32(
                       false, a, false, b1, (short)0, acc1, false, false);
        }
    }
#undef STAGE_X
#undef STAGE_P

    // Per-token sum of squares: lane l covers K%4 in {0,1}, lane l+16 in {2,3}
    float sstot = ss + __shfl_xor(ss, 16, 32);
    float rms   = sqrtf(sstot * (1.0f / NC_) + EPS_);

    // Transpose D tiles through LDS: D layout is col = lane&15, row = r + 8*half
#pragma unroll
    for (int r = 0; r < 8; ++r) {
        int m = r + half * 8;
        Lg[m][n]      = acc0[r];
        Lg[m][n + 16] = acc1[r];
    }
    if (half == 0) rmsS[n] = rms;
    __syncthreads();

    if (lane < 16) {
        const int   m   = lane;
        const float inv = 1.0f / rmsS[m];
        const float ap  = alpha_pre[idx];
        const float aq  = alpha_post[idx];
        const float ar  = alpha_res[idx];

        float Hp[4], Hq[4], Mm[16];
#pragma unroll
        for (int j = 0; j < 4; ++j) {
            float z1 = ap * (Lg[m][j]     * inv) + b_pre [idx * 4 + j];
            float z2 = aq * (Lg[m][4 + j] * inv) + b_post[idx * 4 + j];
            Hp[j] = 1.0f / (1.0f + __expf(-z1));
            Hq[j] = 2.0f / (1.0f + __expf(-z2));
        }
#pragma unroll
        for (int e = 0; e < 16; ++e) {
            float z = ar * (Lg[m][8 + e] * inv) + b_res[idx * 16 + e];
            Mm[e] = __expf(z);
        }
        // Sinkhorn-Knopp: 20 x (row normalize, col normalize) on 4x4
        for (int it = 0; it < SINK_ITERS; ++it) {
#pragma unroll
            for (int i = 0; i < 4; ++i) {
                float s = Mm[i*4] + Mm[i*4+1] + Mm[i*4+2] + Mm[i*4+3];
                float r = 1.0f / s;
                Mm[i*4] *= r; Mm[i*4+1] *= r; Mm[i*4+2] *= r; Mm[i*4+3] *= r;
            }
#pragma unroll
            for (int j = 0; j < 4; ++j) {
                float s = Mm[j] + Mm[4+j] + Mm[8+j] + Mm[12+j];
                float r = 1.0f / s;
                Mm[j] *= r; Mm[4+j] *= r; Mm[8+j] *= r; Mm[12+j] *= r;
            }
        }
        float* cf = coef + (size_t)(b * T_ + t0 + m) * 24;
#pragma unroll
        for (int j = 0; j < 4; ++j) { cf[j] = Hp[j]; cf[4 + j] = Hq[j]; }
#pragma unroll
        for (int e = 0; e < 16; ++e) cf[8 + e] = Mm[e];
    }
}

// ---------------------------------------------------------------------------
// Kernel 3: out[i,c] = sum_j (M[i,j] + H_post[i]*H_pre[j]) * x[j,c]
// One 256-thread block per token; float4-vectorized, fully coalesced,
// x loaded once into registers and reused for all 4 outputs.
// ---------------------------------------------------------------------------
__global__ void __launch_bounds__(256)
combine_kernel(const float* __restrict__ x,
               const float* __restrict__ coef,
               float* __restrict__ out) {
    const int tok = blockIdx.x;             // 0 .. B_*T_ - 1
    __shared__ float cf[24];
    if (threadIdx.x < 24) cf[threadIdx.x] = coef[(size_t)tok * 24 + threadIdx.x];
    __syncthreads();

    const int c0 = threadIdx.x * 4;         // 256 threads * 4 = 1024 = C_
    const float* xr = x + (size_t)tok * NC_;
    float4 xv[4];
#pragma unroll
    for (int j = 0; j < 4; ++j) xv[j] = *(const float4*)(xr + j * C_ + c0);

    float* orow = out + (size_t)tok * NC_;
#pragma unroll
    for (int i = 0; i < 4; ++i) {
        float4 o = make_float4(0.f, 0.f, 0.f, 0.f);
#pragma unroll
        for (int j = 0; j < 4; ++j) {
            float wij = cf[8 + i * 4 + j] + cf[4 + i] * cf[j];
            o.x += wij * xv[j].x;  o.y += wij * xv[j].y;
            o.z += wij * xv[j].z;  o.w += wij * xv[j].w;
        }
        *(float4*)(orow + i * C_ + c0) = o;
    }
}

// ---------------------------------------------------------------------------
extern "C" void kernel_launch(void* const* d_in, const int* in_sizes, int n_in,
                              void* d_out, int out_size, void* d_ws, size_t ws_size,
                              hipStream_t stream) {
    const float* x     = (const float*)d_in[0];
    const float* wgt   = (const float*)d_in[1];
    const float* ppre  = (const float*)d_in[2];
    const float* ppost = (const float*)d_in[3];
    const float* pres  = (const float*)d_in[4];
    const float* bpre  = (const float*)d_in[5];
    const float* bpost = (const float*)d_in[6];
    const float* bres  = (const float*)d_in[7];
    const float* apre  = (const float*)d_in[8];
    const float* apost = (const float*)d_in[9];
    const float* ares  = (const float*)d_in[10];
    const int*   aidx  = (const int*)d_in[11];
    float* out = (float*)d_out;

    // Workspace: packed phi (A*NC*32 floats = 4 MiB) + coefs (B*T*24 floats)
    float* phiP = (float*)d_ws;
    float* coef = phiP + (size_t)A_ * NC_ * JPACK;

    const int packN = A_ * NC_ * JPACK;
    pack_phi_kernel<<<(packN + 255) / 256, 256, 0, stream>>>(ppre, ppost, pres, phiP);

    coef_kernel<<<(B_ * T_) / 16, 32, 0, stream>>>(
        x, wgt, phiP, bpre, bpost, bres, apre, apost, ares, aidx, coef);

    combine_kernel<<<B_ * T_, 256, 0, stream>>>(x, coef, out);
}